// Explainer_72069551227425
// MI455X (gfx1250) — compile-verified
//
#include <hip/hip_runtime.h>

typedef float v2f __attribute__((ext_vector_type(2)));
typedef float v8f __attribute__((ext_vector_type(8)));

#define N_NODES  50000
#define N_EDGES  800000
#define N_GRAPHS 256
#define BN_EPS   1e-5f
#define TEMP     5.0f
#define MT       5          // M-tiles (of 16 rows) per wave; 3125 = 5 * 625

// ---------------------------------------------------------------------------
// agg = xin   (vectorized copy; rows are 16B aligned: din in {128,64})
// ---------------------------------------------------------------------------
__global__ void copy_vec4(const float4* __restrict__ src, float4* __restrict__ dst, int n4) {
    int i = blockIdx.x * blockDim.x + threadIdx.x;
    if (i < n4) dst[i] = src[i];
}

// ---------------------------------------------------------------------------
// agg[dst] += xin[src]  (per-edge scatter-add; one thread per 4 features).
// All of xin (<=25.6MB) lives in the 192MB L2, so gathers + atomics stay
// on-chip; this is the true bottleneck phase (~410MB of L2 gather @ layer 0).
// ---------------------------------------------------------------------------
__global__ void edge_scatter_add(const float* __restrict__ x, const int* __restrict__ src,
                                 const int* __restrict__ dst, float* __restrict__ agg,
                                 int din_q /* din/4 */) {
    long long i = (long long)blockIdx.x * blockDim.x + threadIdx.x;
    long long total = (long long)N_EDGES * din_q;
    if (i >= total) return;
    int e = (int)(i / din_q);
    int c = (int)(i % din_q) * 4;
    int din = din_q * 4;
    int s = src[e], d = dst[e];
    const float4 v = *(const float4*)(x + (long long)s * din + c);
    float* o = agg + (long long)d * din + c;
    atomicAdd(o + 0, v.x);
    atomicAdd(o + 1, v.y);
    atomicAdd(o + 2, v.z);
    atomicAdd(o + 3, v.w);
}

// ---------------------------------------------------------------------------
// Fused GEMM + bias (+BN eval) (+ReLU):  out[M,Nc] = epi(A[M,K] @ B[K,Nc])
//
// One wave owns one 16-column panel of B for the FULL K (K/4 v2f fragments
// preloaded into registers, K is a template constant so the array unrolls),
// then sweeps MT=5 M-tiles of 16 rows, reusing the B panel. Inner loop is
// pure global_load_b64(A-frag) + v_wmma_f32_16x16x4_f32 accumulate.
//
// Fragment layouts (ISA 7.12.2, wave32):
//   A 16x4 f32 : lanes 0-15 -> M=lane,   VGPR{0,1}=K{0,1};
//                lanes 16-31 -> M=lane-16, VGPR{0,1}=K{2,3}.
//   B 4x16     : mirrored (row K striped across lanes; lane-halves = K-halves)
//   C/D 16x16  : VGPR v: lanes 0-15 -> M=v, lanes 16-31 -> M=v+8; N=lane%16.
// M=50000=16*3125, Nc,K multiples of 16/4 -> no tails, EXEC all-1s for WMMA.
// ---------------------------------------------------------------------------
template <int K>
__global__ void wmma_gemm_fused(const float* __restrict__ A, const float* __restrict__ B,
                                const float* __restrict__ bias,
                                const float* __restrict__ gamma, const float* __restrict__ beta,
                                const float* __restrict__ mean,  const float* __restrict__ var,
                                float* __restrict__ out,
                                int Nc, int fuse_bn, int fuse_relu) {
    constexpr int KF = K / 4;
    const int lane  = threadIdx.x;            // 0..31
    const int half  = lane >> 4;              // 0 or 1
    const int l16   = lane & 15;
    const int kh    = half * 2;               // 0 or 2
    const int ntile = blockIdx.y * blockDim.y + threadIdx.y;
    const int col   = ntile * 16 + l16;
    const int m0    = blockIdx.x * (16 * MT);

    // ---- preload full-K B panel into registers (reused across MT m-tiles) --
    v2f bfrag[KF];
#pragma unroll
    for (int kf = 0; kf < KF; ++kf) {
        const float* bp = B + (long long)(kf * 4 + kh) * Nc + col;
        bfrag[kf].x = bp[0];
        bfrag[kf].y = bp[Nc];
    }

    // ---- epilogue constants (BN folded into scale+shift) ------------------
    float bv = bias[col];
    float sc = 1.0f, sh = 0.0f;
    if (fuse_bn) {
        float inv = rsqrtf(var[col] + BN_EPS) * gamma[col];
        sc = inv;
        sh = beta[col] - mean[col] * inv;
    }

    for (int t = 0; t < MT; ++t) {
        const int mbase = m0 + t * 16;
        const float* ap = A + (long long)(mbase + l16) * K + kh;
        v8f acc = {};
#pragma unroll
        for (int kf = 0; kf < KF; ++kf) {
            v2f a;
            a.x = ap[kf * 4];
            a.y = ap[kf * 4 + 1];
            acc = __builtin_amdgcn_wmma_f32_16x16x4_f32(
                /*neg_a=*/false, a, /*neg_b=*/false, bfrag[kf],
                /*c_mod=*/(short)0, acc, /*reuse_a=*/false, /*reuse_b=*/false);
        }
#pragma unroll
        for (int v = 0; v < 8; ++v) {
            int row = mbase + v + half * 8;
            float h = acc[v] + bv;
            h = h * sc + sh;
            if (fuse_relu) h = fmaxf(h, 0.0f);
            out[(long long)row * Nc + col] = h;
        }
    }
}

static inline void launch_gemm(int K, const float* A, const float* B, const float* bias,
                               const float* gm, const float* bt, const float* mn, const float* vr,
                               float* out, int Nc, int fuse_bn, int fuse_relu,
                               hipStream_t stream) {
    int ntiles = Nc / 16;
    int wy = ntiles < 4 ? ntiles : 4;         // waves per block (1 n-tile each)
    dim3 grid(N_NODES / (16 * MT), ntiles / wy);
    dim3 block(32, wy);
    switch (K) {
        case 128: wmma_gemm_fused<128><<<grid, block, 0, stream>>>(A, B, bias, gm, bt, mn, vr, out, Nc, fuse_bn, fuse_relu); break;
        case 64:  wmma_gemm_fused<64> <<<grid, block, 0, stream>>>(A, B, bias, gm, bt, mn, vr, out, Nc, fuse_bn, fuse_relu); break;
        default:  wmma_gemm_fused<32> <<<grid, block, 0, stream>>>(A, B, bias, gm, bt, mn, vr, out, Nc, fuse_bn, fuse_relu); break;
    }
}

// ---------------------------------------------------------------------------
// Segment softmax tail
// ---------------------------------------------------------------------------
__device__ __forceinline__ unsigned f2ord(float f) {
    unsigned u = __float_as_uint(f);
    return (u & 0x80000000u) ? ~u : (u | 0x80000000u);   // order-preserving map
}
__device__ __forceinline__ float ord2f(unsigned u) {
    return (u & 0x80000000u) ? __uint_as_float(u & 0x7fffffffu) : __uint_as_float(~u);
}

__global__ void init_graph_buffers(unsigned* __restrict__ gmax, float* __restrict__ gsum) {
    int i = blockIdx.x * blockDim.x + threadIdx.x;
    if (i < N_GRAPHS) { gmax[i] = 0u; /* == -FLT_MAX under f2ord */ gsum[i] = 0.0f; }
}

__global__ void logits_segmax(const float* __restrict__ x, const float* __restrict__ Wl,
                              const float* __restrict__ bl, const int* __restrict__ batch,
                              float* __restrict__ logits, unsigned* __restrict__ gmax) {
    int n = blockIdx.x * blockDim.x + threadIdx.x;
    if (n >= N_NODES) return;
    float acc = bl[0];
    const float* xr = x + (long long)n * 32;
#pragma unroll
    for (int k = 0; k < 32; ++k) acc += xr[k] * Wl[k];
    float lg = acc * (1.0f / TEMP);
    logits[n] = lg;
    atomicMax(&gmax[batch[n]], f2ord(lg));
}

__global__ void exp_segsum(const float* __restrict__ logits, const int* __restrict__ batch,
                           const unsigned* __restrict__ gmax, float* __restrict__ e_out,
                           float* __restrict__ gsum) {
    int n = blockIdx.x * blockDim.x + threadIdx.x;
    if (n >= N_NODES) return;
    int b = batch[n];
    float e = __expf(logits[n] - ord2f(gmax[b]));
    e_out[n] = e;
    atomicAdd(&gsum[b], e);
}

__global__ void normalize_out(float* __restrict__ e_out, const int* __restrict__ batch,
                              const float* __restrict__ gsum) {
    int n = blockIdx.x * blockDim.x + threadIdx.x;
    if (n >= N_NODES) return;
    e_out[n] = e_out[n] / gsum[batch[n]];
}

// ---------------------------------------------------------------------------
// Host-side launch sequence (all on `stream`, graph-capture safe)
// ---------------------------------------------------------------------------
extern "C" void kernel_launch(void* const* d_in, const int* in_sizes, int n_in,
                              void* d_out, int out_size, void* d_ws, size_t ws_size,
                              hipStream_t stream) {
    (void)in_sizes; (void)n_in; (void)out_size; (void)ws_size;

    const float* x     = (const float*)d_in[0];
    const int*   ei    = (const int*)d_in[1];
    const int*   batch = (const int*)d_in[2];
    const int*   src   = ei;             // edge_index[0]
    const int*   dst   = ei + N_EDGES;   // edge_index[1]
    const float* Wlin  = (const float*)d_in[27];
    const float* blin  = (const float*)d_in[28];

    // Workspace layout (needs 3*25.6MB + N + 2*G floats)
    char*  ws     = (char*)d_ws;
    float* agg    = (float*)ws;                                   // [N,128] max
    float* h1     = (float*)(ws + 1ull * N_NODES * 128 * 4);      // [N,128] max
    float* cur    = (float*)(ws + 2ull * N_NODES * 128 * 4);      // [N,128] max
    float* logits = (float*)(ws + 3ull * N_NODES * 128 * 4);      // [N]
    unsigned* gmax = (unsigned*)(logits + N_NODES);               // [G]
    float*    gsum = (float*)(gmax + N_GRAPHS);                   // [G]

    const int dims[4] = {128, 128, 64, 32};
    const float* xin = x;

    for (int i = 0; i < 3; ++i) {
        const int din = dims[i], dout = dims[i + 1];
        const float* W1 = (const float*)d_in[3 + i * 8 + 0];
        const float* b1 = (const float*)d_in[3 + i * 8 + 1];
        const float* W2 = (const float*)d_in[3 + i * 8 + 2];
        const float* b2 = (const float*)d_in[3 + i * 8 + 3];
        const float* gm = (const float*)d_in[3 + i * 8 + 4];
        const float* bt = (const float*)d_in[3 + i * 8 + 5];
        const float* mn = (const float*)d_in[3 + i * 8 + 6];
        const float* vr = (const float*)d_in[3 + i * 8 + 7];

        // agg = xin + segment_sum(xin[src] -> dst)
        int n4 = N_NODES * din / 4;
        copy_vec4<<<(n4 + 255) / 256, 256, 0, stream>>>((const float4*)xin, (float4*)agg, n4);
        long long tot = (long long)N_EDGES * (din / 4);
        edge_scatter_add<<<(unsigned)((tot + 255) / 256), 256, 0, stream>>>(xin, src, dst, agg, din / 4);

        // h1 = relu(agg @ W1 + b1)
        launch_gemm(din, agg, W1, b1, nullptr, nullptr, nullptr, nullptr,
                    h1, dout, /*bn=*/0, /*relu=*/1, stream);
        // cur = BN(h1 @ W2 + b2)  (+ relu for layers 0,1)
        launch_gemm(dout, h1, W2, b2, gm, bt, mn, vr,
                    cur, dout, /*bn=*/1, /*relu=*/(i < 2) ? 1 : 0, stream);
        xin = cur;
    }

    // logits = (x3 @ W_lin + b_lin)/TEMP ; per-graph softmax over nodes
    init_graph_buffers<<<1, 256, 0, stream>>>(gmax, gsum);
    logits_segmax<<<(N_NODES + 255) / 256, 256, 0, stream>>>(xin, Wlin, blin, batch, logits, gmax);
    exp_segsum<<<(N_NODES + 255) / 256, 256, 0, stream>>>(logits, batch, gmax, (float*)d_out, gsum);
    normalize_out<<<(N_NODES + 255) / 256, 256, 0, stream>>>((float*)d_out, batch, gsum);
}